// LJ_8718783611256
// MI455X (gfx1250) — compile-verified
//
#include <hip/hip_runtime.h>

typedef __attribute__((ext_vector_type(2))) float v2f;
typedef __attribute__((ext_vector_type(8))) float v8f;

#define NPART 2048
#define NTILE 128   // NPART / 16
#define NBATCH 4

// One block = one (batch b, row-tile it). 128 threads = 4 wave32s.
// Each wave walks column tiles jt = wave, wave+4, ... computing a 16x16 Gram
// tile with V_WMMA_F32_16X16X4_F32 and evaluating LJ on the 8 accumulator
// elements it owns. sq_ij = |xi|^2 + |xj|^2 - 2*G_ij  (norms cached in LDS).
__global__ __launch_bounds__(128) void lj_tile_kernel(const float* __restrict__ x,
                                                      float* __restrict__ partial) {
  const int b  = blockIdx.x / NTILE;
  const int it = blockIdx.x % NTILE;
  const float* xb = x + (size_t)b * NPART * 3;

  __shared__ float s_norm[NPART];  // 8 KB of the 320 KB WGP LDS
  for (int r = threadIdx.x; r < NPART; r += 128) {
    float v0 = xb[3 * r + 0], v1 = xb[3 * r + 1], v2 = xb[3 * r + 2];
    s_norm[r] = v0 * v0 + v1 * v1 + v2 * v2;
  }
  __syncthreads();

  const int  lane = threadIdx.x & 31;
  const int  wave = threadIdx.x >> 5;   // 0..3
  const int  r16  = lane & 15;
  const bool hi   = lane >= 16;         // upper half-wave holds K=2,3

  // A tile (rows it*16 .. it*16+15), ISA 16x4 f32 layout:
  //   VGPR0: K=0 (lanes 0-15) / K=2 (lanes 16-31)
  //   VGPR1: K=1 (lanes 0-15) / K=3 (lanes 16-31, padded zero since D=3)
  const int arow = it * 16 + r16;
  v2f A;
  A.x = xb[3 * arow + (hi ? 2 : 0)];
  A.y = hi ? 0.0f : xb[3 * arow + 1];

  // Row norms this lane needs are loop-invariant: C/D layout gives lane the
  // rows M = mbase..mbase+7 across the 8 accumulator VGPRs.
  const int mbase = it * 16 + (hi ? 8 : 0);
  float nm[8];
#pragma unroll
  for (int e = 0; e < 8; ++e) nm[e] = s_norm[mbase + e];

  float acc = 0.0f;
  for (int jt = wave; jt < NTILE; jt += 4) {
    // B tile (4x16 = rows jt*16.. as columns): same per-lane pattern as A.
    const int brow = jt * 16 + r16;
    v2f Bm;
    Bm.x = xb[3 * brow + (hi ? 2 : 0)];
    Bm.y = hi ? 0.0f : xb[3 * brow + 1];

    v8f c = {0.f, 0.f, 0.f, 0.f, 0.f, 0.f, 0.f, 0.f};
    // D = A x B + 0  ->  G_ij = x_i . x_j   (EXEC is all-1s here)
    c = __builtin_amdgcn_wmma_f32_16x16x4_f32(
        /*neg_a=*/false, A, /*neg_b=*/false, Bm,
        /*c_mod=*/(short)0, c, /*reuse_a=*/false, /*reuse_b=*/false);

    const int   nIdx = jt * 16 + r16;   // this lane's column index
    const float nn   = s_norm[nIdx];
#pragma unroll
    for (int e = 0; e < 8; ++e) {
      const int m  = mbase + e;
      float sq = nm[e] + nn - 2.0f * c[e];   // r^2
      sq = fmaxf(sq, 1e-12f);                // diagonal / cancellation guard
      const float inv = __builtin_amdgcn_rcpf(sq);  // r^-2 (no sqrt needed)
      const float p6  = inv * inv * inv;            // r^-6
      const float pp  = 4.0f * (p6 * p6 - p6);      // LJ, eps=sigma=1
      acc += (m == nIdx) ? 0.0f : pp;               // drop diagonal
    }
  }

  __shared__ float s_red[128];
  s_red[threadIdx.x] = acc;
  __syncthreads();
  for (int s = 64; s > 0; s >>= 1) {
    if (threadIdx.x < s) s_red[threadIdx.x] += s_red[threadIdx.x + s];
    __syncthreads();
  }
  if (threadIdx.x == 0) partial[blockIdx.x] = s_red[0];  // deterministic slot
}

// One block per batch: harmonic term via sum(x), sum(|x|^2):
//   sum|x-com|^2 = sum|x|^2 - |sum x|^2 / N,  harm = 0.25 * (...)  (k=0.5)
// plus reduction of the 128 LJ tile partials; writes -(0.5*pairsum + harm).
__global__ __launch_bounds__(256) void harm_finalize_kernel(const float* __restrict__ x,
                                                            const float* __restrict__ partial,
                                                            float* __restrict__ out) {
  const int b   = blockIdx.x;
  const int tid = threadIdx.x;
  const float* xb = x + (size_t)b * NPART * 3;

  float s0 = 0.f, s1 = 0.f, s2 = 0.f, ssq = 0.f;
  for (int r = tid; r < NPART; r += 256) {
    float v0 = xb[3 * r + 0], v1 = xb[3 * r + 1], v2 = xb[3 * r + 2];
    s0 += v0; s1 += v1; s2 += v2;
    ssq += v0 * v0 + v1 * v1 + v2 * v2;
  }
  float pot = (tid < NTILE) ? partial[b * NTILE + tid] : 0.0f;

  __shared__ float r0[256], r1[256], r2[256], rq[256], rp[256];
  r0[tid] = s0; r1[tid] = s1; r2[tid] = s2; rq[tid] = ssq; rp[tid] = pot;
  __syncthreads();
  for (int s = 128; s > 0; s >>= 1) {
    if (tid < s) {
      r0[tid] += r0[tid + s];
      r1[tid] += r1[tid + s];
      r2[tid] += r2[tid + s];
      rq[tid] += rq[tid + s];
      rp[tid] += rp[tid + s];
    }
    __syncthreads();
  }
  if (tid == 0) {
    const float invN = 1.0f / (float)NPART;
    const float harm =
        0.25f * (rq[0] - (r0[0] * r0[0] + r1[0] * r1[0] + r2[0] * r2[0]) * invN);
    out[b] = -(0.5f * rp[0] + harm);
  }
}

extern "C" void kernel_launch(void* const* d_in, const int* in_sizes, int n_in,
                              void* d_out, int out_size, void* d_ws, size_t ws_size,
                              hipStream_t stream) {
  (void)in_sizes; (void)n_in; (void)out_size; (void)ws_size;
  const float* x   = (const float*)d_in[0];   // (4, 2048, 3) f32
  float*       out = (float*)d_out;           // (4,) f32
  float*       partial = (float*)d_ws;        // NBATCH*NTILE = 512 floats scratch

  lj_tile_kernel<<<dim3(NBATCH * NTILE), dim3(128), 0, stream>>>(x, partial);
  harm_finalize_kernel<<<dim3(NBATCH), dim3(256), 0, stream>>>(x, partial, out);
}